// ChainCRF_53334903881895
// MI455X (gfx1250) — compile-verified
//
#include <hip/hip_runtime.h>
#include <hip/hip_bf16.h>

// ---------------------------------------------------------------------------
// ChainCRF NLL for MI455X (gfx1250, wave32).
//  Kernel 1: feats = hidden @ W^T + b  via V_WMMA_F32_16X16X4_F32 (fp32 WMMA).
//  Kernel 2: CRF forward scan, one wave per batch, exp(trans) mat-vec trick.
// ---------------------------------------------------------------------------

typedef __attribute__((ext_vector_type(2))) float v2f;
typedef __attribute__((ext_vector_type(8))) float v8f;

#define BATCH 128
#define TLEN  1024
#define HDIM  512
#define KTAGS 52
#define NPAD  64          // feats row stride in workspace
#define MBLK  64          // rows of hidden per workgroup (4 x 16 sub-tiles)
#define KCHNK 64          // k-chunk staged in LDS per iteration
#define WSTRIDE 516       // 512 + 4 pad (avoid 64-bank conflicts)
#define ASTRIDE 68        // 64 + 4 pad
#define ROOT_TAG 0
#define END_TAG  1

// ------------------------------ GEMM kernel --------------------------------
__global__ __launch_bounds__(128)
void feats_gemm_kernel(const float* __restrict__ hidden,   // [B*T, 512]
                       const float* __restrict__ Wm,       // [52, 512]
                       const float* __restrict__ bias,     // [52]
                       float* __restrict__ feats)          // [B*T, 64] (ws)
{
    __shared__ float wlds[KTAGS * WSTRIDE];   // 107 KB, W resident whole kernel
    __shared__ float alds[MBLK * ASTRIDE];    // 17 KB, A k-chunk

    const int tid  = threadIdx.x;
    const int wave = tid >> 5;                // 4 waves: N-tiles 0..3
    const int lane = tid & 31;
    const int ln   = lane & 15;
    const int koff = (lane < 16) ? 0 : 2;     // K half per lane group (ISA layout)
    const int m0   = blockIdx.x * MBLK;

    // Stage W (row-major, padded stride). Row i: thread tid loads float4 col tid*4.
    for (int i = 0; i < KTAGS; ++i) {
        const float4 v = *(const float4*)(Wm + (size_t)i * HDIM + tid * 4);
        *(float4*)(wlds + i * WSTRIDE + tid * 4) = v;
    }

    v8f acc[4] = {};                          // 4 M-subtiles of 16x16 f32

    for (int kc = 0; kc < HDIM; kc += KCHNK) {
        __syncthreads();                      // alds reuse guard
        // Stage A chunk: 64 rows x 64 floats, fully coalesced float4 loads.
        #pragma unroll
        for (int i = 0; i < 8; ++i) {
            const int idx = i * 128 + tid;
            const int r   = idx >> 4;         // 16 float4 per row
            const int c4  = idx & 15;
            const float4 v = *(const float4*)(hidden + (size_t)(m0 + r) * HDIM + kc + c4 * 4);
            *(float4*)(alds + r * ASTRIDE + c4 * 4) = v;
        }
        __syncthreads();

        // 16 WMMA k-steps of depth 4 per chunk.
        for (int kk = 0; kk < KCHNK; kk += 4) {
            // B fragment (4x16 of W^T): lane ln -> N = wave*16+ln, two K values.
            const v2f bfrag = *(const v2f*)(wlds + (wave * 16 + ln) * WSTRIDE + kc + kk + koff);
            #pragma unroll
            for (int ms = 0; ms < 4; ++ms) {
                // A fragment (16x4): lane ln -> M = ms*16+ln, two K values.
                const v2f afrag = *(const v2f*)(alds + (ms * 16 + ln) * ASTRIDE + kk + koff);
                acc[ms] = __builtin_amdgcn_wmma_f32_16x16x4_f32(
                    false, afrag, false, bfrag, (short)0, acc[ms], false, false);
            }
        }
    }

    // Epilogue: C/D layout -> VGPR r: lanes 0-15 row M=r, lanes 16-31 row M=r+8.
    const int n = wave * 16 + ln;
    if (n < KTAGS) {
        const float bv = bias[n];
        #pragma unroll
        for (int ms = 0; ms < 4; ++ms) {
            #pragma unroll
            for (int r = 0; r < 8; ++r) {
                const int m = m0 + ms * 16 + r + ((lane < 16) ? 0 : 8);
                feats[(size_t)m * NPAD + n] = acc[ms][r] + bv;
            }
        }
    }
}

// ------------------------------ scan kernel --------------------------------
__device__ __forceinline__ float wave_max32(float x) {
    #pragma unroll
    for (int o = 16; o > 0; o >>= 1) x = fmaxf(x, __shfl_xor(x, o, 32));
    return x;
}
__device__ __forceinline__ float wave_sum32(float x) {
    #pragma unroll
    for (int o = 16; o > 0; o >>= 1) x += __shfl_xor(x, o, 32);
    return x;
}

__global__ __launch_bounds__(32)
void crf_scan_kernel(const float* __restrict__ feats,     // [B*T, 64] (ws)
                     const float* __restrict__ trans,     // [52, 52]
                     const int*   __restrict__ tags,      // [B, T]
                     const int*   __restrict__ lengths,   // [B]
                     float* __restrict__ out)             // [B]
{
    __shared__ float tlds[KTAGS * KTAGS];

    const int b    = blockIdx.x;
    const int lane = threadIdx.x;
    const int k0   = lane;            // state 0..31 (always valid)
    const int k1   = lane + 32;       // state 32..63
    const bool v1  = (k1 < KTAGS);

    for (int i = lane; i < KTAGS * KTAGS; i += 32) tlds[i] = trans[i];
    __syncthreads();

    // E = exp(trans), two rows per lane, register resident (~104 VGPRs).
    float E0[KTAGS], E1[KTAGS];
    #pragma unroll
    for (int j = 0; j < KTAGS; ++j) {
        E0[j] = __expf(tlds[k0 * KTAGS + j]);            // exp(-1e4) -> exact 0
        E1[j] = v1 ? __expf(tlds[k1 * KTAGS + j]) : 0.0f;
    }

    const int len = lengths[b];
    const float* fb = feats + (size_t)b * TLEN * NPAD;

    float alpha0 = (k0 == ROOT_TAG) ? 0.0f : -1000.0f;
    float alpha1 = v1 ? -1000.0f : -1e30f;

    const int steps = len - 1;        // reference updates only t < len-1
    for (int t = 0; t < steps; ++t) {
        const float f0 = fb[t * NPAD + k0];
        const float f1 = v1 ? fb[t * NPAD + k1] : 0.0f;

        const float m  = wave_max32(fmaxf(alpha0, alpha1));
        const float p0 = __expf(alpha0 - m);
        const float p1 = __expf(alpha1 - m);

        // s[k] = sum_j E[k,j] * p[j]; p broadcast via readlane, 4-way ILP.
        float s0a = 0.0f, s0b = 0.0f, s1a = 0.0f, s1b = 0.0f;
        #pragma unroll
        for (int j = 0; j < 32; j += 2) {
            const float pa = __shfl(p0, j, 32);
            const float pb = __shfl(p0, j + 1, 32);
            s0a = fmaf(E0[j],     pa, s0a);  s1a = fmaf(E1[j],     pa, s1a);
            s0b = fmaf(E0[j + 1], pb, s0b);  s1b = fmaf(E1[j + 1], pb, s1b);
        }
        #pragma unroll
        for (int j = 0; j < KTAGS - 32; j += 2) {        // sources 32..51
            const float pa = __shfl(p1, j, 32);
            const float pb = __shfl(p1, j + 1, 32);
            s0a = fmaf(E0[32 + j],     pa, s0a);  s1a = fmaf(E1[32 + j],     pa, s1a);
            s0b = fmaf(E0[32 + j + 1], pb, s0b);  s1b = fmaf(E1[32 + j + 1], pb, s1b);
        }
        alpha0 = f0 + m + __logf(s0a + s0b);
        alpha1 = v1 ? (f1 + m + __logf(s1a + s1b)) : -1e30f;
    }

    // log_z = logsumexp_k(alpha[k] + trans[END, k])
    const float z0 = alpha0 + tlds[END_TAG * KTAGS + k0];
    const float z1 = v1 ? (alpha1 + tlds[END_TAG * KTAGS + k1]) : -1e30f;
    const float mz = wave_max32(fmaxf(z0, z1));
    const float se = wave_sum32(__expf(z0 - mz) + (v1 ? __expf(z1 - mz) : 0.0f));
    const float log_z = mz + __logf(se);

    // Gold path score: sum over t < len of trans[tag_t, prev] + emit (not last).
    const int* tg = tags + (size_t)b * TLEN;
    float g = 0.0f;
    for (int t = lane; t < len; t += 32) {
        const int tagt = tg[t];
        const int prev = (t == 0) ? ROOT_TAG : tg[t - 1];
        const float tr = tlds[tagt * KTAGS + prev];
        const float em = (t == len - 1) ? 0.0f : fb[t * NPAD + tagt];
        g += tr + em;
    }
    g = wave_sum32(g);

    if (lane == 0) out[b] = log_z - g;
}

// ------------------------------ launcher -----------------------------------
extern "C" void kernel_launch(void* const* d_in, const int* in_sizes, int n_in,
                              void* d_out, int out_size, void* d_ws, size_t ws_size,
                              hipStream_t stream) {
    const float* hidden  = (const float*)d_in[0];   // [B, T, H]
    const float* Wm      = (const float*)d_in[1];   // [K, H]
    const float* bias    = (const float*)d_in[2];   // [K]
    const float* trans   = (const float*)d_in[3];   // [K, K]
    const int*   tags    = (const int*)d_in[4];     // [B, T]
    const int*   lengths = (const int*)d_in[5];     // [B]
    float*       out     = (float*)d_out;           // [B]
    float*       feats   = (float*)d_ws;            // [B*T, 64] = 33.5 MB scratch

    feats_gemm_kernel<<<(BATCH * TLEN) / MBLK, 128, 0, stream>>>(hidden, Wm, bias, feats);
    crf_scan_kernel<<<BATCH, 32, 0, stream>>>(feats, trans, tags, lengths, out);
}